// Net_13134009991724
// MI455X (gfx1250) — compile-verified
//
#include <hip/hip_runtime.h>
#include <math.h>

// Problem constants (from reference setup_inputs)
#define CN0   50000
#define CN1   10000
#define CN2   2048
#define DIN   256
#define DHID  1000
#define NCLS  5000
#define CE0   500000
#define CE1   100000

typedef float v2f __attribute__((ext_vector_type(2)));
typedef float v8f __attribute__((ext_vector_type(8)));

// -------------------- utility kernels --------------------

__global__ void zero_f32(float* __restrict__ p, long n) {
    long i = (long)blockIdx.x * blockDim.x + threadIdx.x;
    if (i < n) p[i] = 0.0f;
}

// deg_out over src, deg_in over dst (counts as f32)
__global__ void degree_kernel(const int* __restrict__ src, const int* __restrict__ dst,
                              float* __restrict__ degs, float* __restrict__ degd, int E) {
    int e = blockIdx.x * blockDim.x + threadIdx.x;
    if (e >= E) return;
    atomicAdd(&degs[src[e]], 1.0f);
    atomicAdd(&degd[dst[e]], 1.0f);
}

// one thread per (edge, 4-feature chunk): agg[dst] += x[src] * rsqrt(max(deg_out[src],1))
__global__ void scatter_kernel(const float* __restrict__ x, const int* __restrict__ src,
                               const int* __restrict__ dst, const float* __restrict__ degout,
                               float* __restrict__ agg, int E, int F) {
    const int chunks = F >> 2;
    long idx = (long)blockIdx.x * blockDim.x + threadIdx.x;
    long total = (long)E * chunks;
    if (idx >= total) return;
    int e = (int)(idx / chunks);
    int c = (int)(idx - (long)e * chunks);
    int s = src[e];
    int d = dst[e];
    float sc = rsqrtf(fmaxf(degout[s], 1.0f));
    const float4 v = *(const float4*)(x + (long)s * F + (long)c * 4);
    float* base = agg + (long)d * F + (long)c * 4;
    atomicAdd(base + 0, v.x * sc);
    atomicAdd(base + 1, v.y * sc);
    atomicAdd(base + 2, v.z * sc);
    atomicAdd(base + 3, v.w * sc);
}

// a[r, :] *= rsqrt(max(deg_in[r],1))
__global__ void rowscale_kernel(float* __restrict__ a, const float* __restrict__ deg,
                                long n, int F) {
    long i = (long)blockIdx.x * blockDim.x + threadIdx.x;
    if (i >= n) return;
    int r = (int)(i / F);
    a[i] *= rsqrtf(fmaxf(deg[r], 1.0f));
}

// -------------------- fp32 WMMA GEMM --------------------
// C[M,N] = A[M,K] @ B[K,N] + bias ; optional sigmoid.
// Block: 256 threads = 8 waves. Block tile 128x64, each wave owns 16 rows x 64 cols.
// K staged via LDS (BK=32), software-pipelined: next tile prefetched to registers
// while current tile is consumed.
//
// V_WMMA_F32_16X16X4_F32 VGPR striping:
//   A 16x4:  lanes 0-15 hold K={k0,k0+1}, lanes 16-31 hold K={k0+2,k0+3}
//   B 4x16:  lanes 0-15 (N=n) hold K={k0,k0+1}, lanes 16-31 hold K={k0+2,k0+3}
//   C 16x16: vgpr r: lanes 0-15 -> M=r, lanes 16-31 -> M=r+8
//
// B is stored in LDS as K-pairs Bs[kp][n][2] so each B fragment is one aligned
// ds_load_b64 (no packing movs). Row pad +16 => row stride 160 dwords == 32 mod 64,
// so the two lane-halves (kp vs kp+1) use disjoint bank halves.

#define BM 128
#define BN 64
#define BK 32
#define BNP (BN + 16)

__global__ void __launch_bounds__(256)
gemm_wmma_f32(const float* __restrict__ A, const float* __restrict__ B,
              const float* __restrict__ bias, float* __restrict__ C,
              int M, int N, int K, int act) {
    __shared__ float As[BM][BK + 4];       // +4 pad: aligned float4 stores, conflict-free b64 reads
    __shared__ float Bs[BK / 2][BNP][2];   // fragment-ready K-pair layout

    const int tid  = threadIdx.x;
    const int lane = tid & 31;
    const int wave = tid >> 5;
    const int hl   = lane & 15;
    const bool hi  = lane >= 16;
    const int blockRow = blockIdx.y * BM;
    const int blockCol = blockIdx.x * BN;
    const int wrow = wave * 16;

    float4 aReg[4];
    float4 bReg[2];

    auto load_tile = [&](int k0) {
#pragma unroll
        for (int i = 0; i < 4; ++i) {                 // A: 128x32 = 1024 float4, 4/thread
            int c = tid + i * 256;
            int r = c >> 3, kc = (c & 7) << 2;
            int grow = blockRow + r, gk = k0 + kc;
            float4 v = make_float4(0.f, 0.f, 0.f, 0.f);
            if (grow < M && (gk + 3) < K)             // K % 4 == 0 -> chunks never straddle
                v = *(const float4*)(A + (long)grow * K + gk);
            aReg[i] = v;
        }
#pragma unroll
        for (int i = 0; i < 2; ++i) {                 // B: 32x64 = 512 float4, 2/thread
            int c = tid + i * 256;
            int r = c >> 4, nc = (c & 15) << 2;
            int gk = k0 + r, gcol = blockCol + nc;
            float4 v = make_float4(0.f, 0.f, 0.f, 0.f);
            if (gk < K && (gcol + 3) < N)             // N % 4 == 0
                v = *(const float4*)(B + (long)gk * N + gcol);
            bReg[i] = v;
        }
    };

    auto store_tile = [&]() {
#pragma unroll
        for (int i = 0; i < 4; ++i) {
            int c = tid + i * 256;
            int r = c >> 3, kc = (c & 7) << 2;
            *(float4*)&As[r][kc] = aReg[i];
        }
#pragma unroll
        for (int i = 0; i < 2; ++i) {
            int c = tid + i * 256;
            int r = c >> 4, nc = (c & 15) << 2;
            int kp = r >> 1, jj = r & 1;
            Bs[kp][nc + 0][jj] = bReg[i].x;
            Bs[kp][nc + 1][jj] = bReg[i].y;
            Bs[kp][nc + 2][jj] = bReg[i].z;
            Bs[kp][nc + 3][jj] = bReg[i].w;
        }
    };

    v8f acc[4];
    const v8f zero8 = {};
#pragma unroll
    for (int j = 0; j < 4; ++j) acc[j] = zero8;

    const int numK = (K + BK - 1) / BK;

    load_tile(0);
    store_tile();
    __syncthreads();

    for (int t = 0; t < numK; ++t) {
        if (t + 1 < numK) load_tile((t + 1) * BK);    // hide global latency behind WMMAs

#pragma unroll
        for (int ks = 0; ks < BK / 4; ++ks) {
            const int kk = ks * 4 + (hi ? 2 : 0);
            const int kp = ks * 2 + (hi ? 1 : 0);
            v2f a = *(const v2f*)&As[wrow + hl][kk];  // aligned b64, conflict-free
#pragma unroll
            for (int j = 0; j < 4; ++j) {
                const int col = j * 16 + hl;
                v2f b = *(const v2f*)&Bs[kp][col][0]; // one aligned b64 per fragment
                acc[j] = __builtin_amdgcn_wmma_f32_16x16x4_f32(
                    /*neg_a=*/false, a, /*neg_b=*/false, b,
                    /*c_mod=*/(short)0, acc[j],
                    /*reuse_a=*/false, /*reuse_b=*/false);
            }
        }
        __syncthreads();                               // all reads of tile t done
        if (t + 1 < numK) {
            store_tile();                              // commit tile t+1
            __syncthreads();
        }
    }

    // Epilogue: bias (+ sigmoid), bounds-checked stores.
#pragma unroll
    for (int j = 0; j < 4; ++j) {
        const int gcol = blockCol + j * 16 + hl;
        if (gcol >= N) continue;
        const float bv = bias[gcol];
#pragma unroll
        for (int r = 0; r < 8; ++r) {
            const int grow = blockRow + wrow + (hi ? 8 : 0) + r;
            if (grow < M) {
                float v = acc[j][r] + bv;
                if (act) v = 1.0f / (1.0f + __expf(-v));
                C[(long)grow * N + gcol] = v;
            }
        }
    }
}

// -------------------- launch --------------------

extern "C" void kernel_launch(void* const* d_in, const int* in_sizes, int n_in,
                              void* d_out, int out_size, void* d_ws, size_t ws_size,
                              hipStream_t stream) {
    const float* x   = (const float*)d_in[0];
    const float* W1  = (const float*)d_in[1];
    const float* b1  = (const float*)d_in[2];
    const float* W2  = (const float*)d_in[3];
    const float* b2  = (const float*)d_in[4];
    const int* e0s   = (const int*)d_in[5];
    const int* e0d   = (const int*)d_in[6];
    const int* e1s   = (const int*)d_in[7];
    const int* e1d   = (const int*)d_in[8];
    float* out = (float*)d_out;

    // Workspace layout (floats): zero-region first so one memclear covers it.
    float* ws       = (float*)d_ws;
    float* deg0_out = ws;                           // CN0
    float* deg0_in  = deg0_out + CN0;               // CN1
    float* deg1_out = deg0_in + CN1;                // CN1
    float* deg1_in  = deg1_out + CN1;               // CN2
    float* agg1     = deg1_in + CN2;                // CN1*DIN  (16B-aligned offset)
    float* agg2     = agg1 + (long)CN1 * DIN;       // CN2*DHID
    float* h1       = agg2 + (long)CN2 * DHID;      // CN1*DHID (fully overwritten by GEMM1)

    const long nzero = (long)(CN0 + 2 * CN1 + CN2) + (long)CN1 * DIN + (long)CN2 * DHID;
    zero_f32<<<(int)((nzero + 255) / 256), 256, 0, stream>>>(ws, nzero);

    degree_kernel<<<(CE0 + 255) / 256, 256, 0, stream>>>(e0s, e0d, deg0_out, deg0_in, CE0);
    degree_kernel<<<(CE1 + 255) / 256, 256, 0, stream>>>(e1s, e1d, deg1_out, deg1_in, CE1);

    // Layer 1 aggregation: agg1[dst] += x[src]/sqrt(deg_out[src])
    const long t1 = (long)CE0 * (DIN / 4);
    scatter_kernel<<<(int)((t1 + 255) / 256), 256, 0, stream>>>(x, e0s, e0d, deg0_out, agg1, CE0, DIN);
    const long na1 = (long)CN1 * DIN;
    rowscale_kernel<<<(int)((na1 + 255) / 256), 256, 0, stream>>>(agg1, deg0_in, na1, DIN);

    // GEMM1: h1[10000,1000] = agg1 @ W1 + b1
    dim3 g1((DHID + BN - 1) / BN, (CN1 + BM - 1) / BM);
    gemm_wmma_f32<<<g1, 256, 0, stream>>>(agg1, W1, b1, h1, CN1, DHID, DIN, 0);

    // Layer 2 aggregation: agg2[dst] += h1[src]/sqrt(deg_out[src])
    const long t2 = (long)CE1 * (DHID / 4);
    scatter_kernel<<<(int)((t2 + 255) / 256), 256, 0, stream>>>(h1, e1s, e1d, deg1_out, agg2, CE1, DHID);
    const long na2 = (long)CN2 * DHID;
    rowscale_kernel<<<(int)((na2 + 255) / 256), 256, 0, stream>>>(agg2, deg1_in, na2, DHID);

    // GEMM2 + sigmoid: out[2048,5000] = sigmoid(agg2 @ W2 + b2)
    dim3 g2((NCLS + BN - 1) / BN, (CN2 + BM - 1) / BM);
    gemm_wmma_f32<<<g2, 256, 0, stream>>>(agg2, W2, b2, out, CN2, NCLS, DHID, 1);
}